// MMGCN_88278757802633
// MI455X (gfx1250) — compile-verified
//
#include <hip/hip_runtime.h>

typedef float v2f __attribute__((ext_vector_type(2)));
typedef float v8f __attribute__((ext_vector_type(8)));

// ---------------------------------------------------------------------------
// Degree / normalization:  deg = 1 (self loop) + sum over edges; dinv = rsqrt
// ---------------------------------------------------------------------------
__global__ void k_deg_init(float* __restrict__ deg, int n) {
    int i = blockIdx.x * blockDim.x + threadIdx.x;
    if (i < n) deg[i] = 1.0f;                       // self-loop contribution
}

__global__ void k_deg_edges(const int* __restrict__ dst, float* __restrict__ deg, int e) {
    int i = blockIdx.x * blockDim.x + threadIdx.x;
    if (i < e) atomicAdd(&deg[dst[i]], 1.0f);
}

__global__ void k_dinv(float* __restrict__ deg, int n) {
    int i = blockIdx.x * blockDim.x + threadIdx.x;
    if (i < n) deg[i] = rsqrtf(deg[i]);             // deg >= 1 always
}

// ---------------------------------------------------------------------------
// Dense GEMM  Y[N,FOUT] = X[N,FIN] @ W[FIN,FOUT]  (+bias, +relu) via
// V_WMMA_F32_16X16X4_F32.  One wave32 per 16x16 C tile, K stepped by 4.
//   A 16x4 f32 layout : lanes 0-15 M=0..15 {v0=K0, v1=K1}; lanes 16-31 {K2,K3}
//   B 4x16 f32 layout : lanes 0-15 N=0..15 {v0=K0, v1=K1}; lanes 16-31 {K2,K3}
//   C 16x16 f32       : vgpr j -> M = j + 8*(lane>=16), N = lane&15
// N rows must be a multiple of 16 (100000 = 6250 * 16).
// ---------------------------------------------------------------------------
template <int FIN, int FOUT, bool BIAS, bool RELU>
__global__ __launch_bounds__(128) void k_gemm_wmma(
    const float* __restrict__ X, const float* __restrict__ W,
    const float* __restrict__ bias, float* __restrict__ Y, int nTilesM)
{
    const int lane   = threadIdx.x & 31;
    const int wave   = threadIdx.x >> 5;
    const int tilesN = FOUT / 16;
    const int tile   = blockIdx.x * 4 + wave;
    if (tile >= nTilesM * tilesN) return;           // wave-uniform exit
    const int tm   = tile / tilesN;
    const int tn   = tile % tilesN;
    const int m    = lane & 15;                     // row (A) / col (B,C)
    const int hi   = lane >> 4;
    const int koff = hi * 2;

    const float* __restrict__ xrow = X + (size_t)(tm * 16 + m) * FIN;
    const float* __restrict__ wcol = W + tn * 16 + m;

    v8f c = {};
#pragma unroll
    for (int k = 0; k < FIN; k += 4) {
        v2f a, b;
        a.x = xrow[k + koff];
        a.y = xrow[k + koff + 1];
        b.x = wcol[(size_t)(k + koff) * FOUT];
        b.y = wcol[(size_t)(k + koff + 1) * FOUT];
        // D = A*B + C  (8-arg form: neg_a, A, neg_b, B, c_mod, C, reuse_a, reuse_b)
        c = __builtin_amdgcn_wmma_f32_16x16x4_f32(false, a, false, b,
                                                  (short)0, c, false, false);
    }

    const int col = tn * 16 + m;
    float bv = 0.0f;
    if (BIAS) bv = bias[col];
#pragma unroll
    for (int j = 0; j < 8; ++j) {
        int r   = tm * 16 + j + 8 * hi;
        float v = c[j] + bv;
        if (RELU) v = fmaxf(v, 0.0f);
        Y[(size_t)r * FOUT + col] = v;
    }
}

// ---------------------------------------------------------------------------
// GCN aggregation.  Out = bias + dinv[i]^2 * H[i]   (self loop term), then
// Out[dst] += dinv[src]*dinv[dst] * H[src] over all edges (atomic scatter).
// fshift = log2(fout) (6 or 7).
// ---------------------------------------------------------------------------
__global__ void k_agg_init(const float* __restrict__ H, const float* __restrict__ bias,
                           const float* __restrict__ dinv, float* __restrict__ Out,
                           int total, int fshift)
{
    int idx = blockIdx.x * blockDim.x + threadIdx.x;
    if (idx >= total) return;
    int i = idx >> fshift;
    int f = idx & ((1 << fshift) - 1);
    float di = dinv[i];
    Out[idx] = bias[f] + H[idx] * di * di;
}

__global__ void k_agg_edges(const float* __restrict__ H, const int* __restrict__ src,
                            const int* __restrict__ dst, const float* __restrict__ dinv,
                            float* __restrict__ Out, int nWork, int fshift)
{
    int idx = blockIdx.x * blockDim.x + threadIdx.x;
    if (idx >= nWork) return;
    const int q = fshift - 2;                       // (fout/4) chunks per edge
    int e = idx >> q;
    int j = (idx & ((1 << q) - 1)) * 4;
    int s = src[e], d = dst[e];
    float cn = dinv[s] * dinv[d];
    const float4 h = *(const float4*)(H + ((size_t)s << fshift) + j);
    float* o = Out + ((size_t)d << fshift) + j;
    atomicAdd(o + 0, h.x * cn);
    atomicAdd(o + 1, h.y * cn);
    atomicAdd(o + 2, h.z * cn);
    atomicAdd(o + 3, h.w * cn);
}

// ---------------------------------------------------------------------------
// zc = (z0 + z1) * 0.5
// ---------------------------------------------------------------------------
__global__ void k_combine(const float* __restrict__ z0, const float* __restrict__ z1,
                          float* __restrict__ zc, int total)
{
    int i = blockIdx.x * blockDim.x + threadIdx.x;
    if (i < total) zc[i] = (z0[i] + z1[i]) * 0.5f;
}

// ---------------------------------------------------------------------------
// Edge decode: out[l] = dot(zc[s_l], zc[d_l]) over 64 feats.
// 16 lanes cooperate per pair; xor-shuffle reduction within the half-wave.
// ---------------------------------------------------------------------------
__global__ void k_decode(const float* __restrict__ zc, const int* __restrict__ sl,
                         const int* __restrict__ dl, float* __restrict__ out, int L)
{
    int idx    = blockIdx.x * blockDim.x + threadIdx.x;
    int pair   = idx >> 4;
    int lane16 = idx & 15;
    if (pair >= L) return;
    int s = sl[pair], d = dl[pair];
    const float* a = zc + (size_t)s * 64;
    const float* b = zc + (size_t)d * 64;
    float sum = 0.0f;
#pragma unroll
    for (int f = lane16; f < 64; f += 16) sum += a[f] * b[f];
    sum += __shfl_xor(sum, 8, 16);
    sum += __shfl_xor(sum, 4, 16);
    sum += __shfl_xor(sum, 2, 16);
    sum += __shfl_xor(sum, 1, 16);
    if (lane16 == 0) out[pair] = sum;
}

// ---------------------------------------------------------------------------
static inline int cdiv_i(long long a, long long b) { return (int)((a + b - 1) / b); }

extern "C" void kernel_launch(void* const* d_in, const int* in_sizes, int n_in,
                              void* d_out, int out_size, void* d_ws, size_t ws_size,
                              hipStream_t stream)
{
    const float* x   = (const float*)d_in[0];
    const int    N   = in_sizes[0] / 128;
    const int    E   = in_sizes[1] / 2;
    const int    L   = in_sizes[2] / 2;
    const int* src = (const int*)d_in[1];
    const int* dst = src + E;
    const int* sl  = (const int*)d_in[2];
    const int* dl  = sl + L;

    const float* Wt[2][4];
    const float* Bt[2][4];
    {
        int k = 3;
        for (int p = 0; p < 2; ++p)
            for (int l = 0; l < 4; ++l) {
                Wt[p][l] = (const float*)d_in[k++];
                Bt[p][l] = (const float*)d_in[k++];
            }
    }

    float* out = (float*)d_out;

    // workspace layout (floats)
    float* ws   = (float*)d_ws;
    float* dinv = ws;                               // [N]
    float* buf0 = dinv + N;                         // [N,128]
    float* buf1 = buf0 + (size_t)N * 128;           // [N,128]
    float* z0   = buf1 + (size_t)N * 128;           // [N,64]
    float* z1   = z0 + (size_t)N * 64;              // [N,64]
    float* zc   = z1 + (size_t)N * 64;              // [N,64]

    const int TB     = 256;
    const int tilesM = N / 16;                      // N = 100000 -> exact

    // ---- degrees / dinv ----
    k_deg_init<<<cdiv_i(N, TB), TB, 0, stream>>>(dinv, N);
    k_deg_edges<<<cdiv_i(E, TB), TB, 0, stream>>>(dst, dinv, E);
    k_dinv<<<cdiv_i(N, TB), TB, 0, stream>>>(dinv, N);

    // ================= Path 0 (hidden = 64) =================
    {
        const int F = 64, SH = 6;
        int gTiles = cdiv_i((long long)tilesM * (F / 16), 4);
        int gInit  = cdiv_i((long long)N * F, TB);
        int gEdge  = cdiv_i((long long)E * (F / 4), TB);

        // conv1: x[128] @ W1 -> buf0 ; aggregate -> buf1
        k_gemm_wmma<128, 64, false, false><<<gTiles, 128, 0, stream>>>(x, Wt[0][0], nullptr, buf0, tilesM);
        k_agg_init<<<gInit, TB, 0, stream>>>(buf0, Bt[0][0], dinv, buf1, N * F, SH);
        k_agg_edges<<<gEdge, TB, 0, stream>>>(buf0, src, dst, dinv, buf1, E * (F / 4), SH);
        // conv2: buf1 @ W2 -> buf0 ; aggregate -> buf1
        k_gemm_wmma<64, 64, false, false><<<gTiles, 128, 0, stream>>>(buf1, Wt[0][1], nullptr, buf0, tilesM);
        k_agg_init<<<gInit, TB, 0, stream>>>(buf0, Bt[0][1], dinv, buf1, N * F, SH);
        k_agg_edges<<<gEdge, TB, 0, stream>>>(buf0, src, dst, dinv, buf1, E * (F / 4), SH);
        // relu(buf1 @ W3 + b3) -> buf0 ; buf0 @ W4 + b4 -> z0
        k_gemm_wmma<64, 64, true, true><<<gTiles, 128, 0, stream>>>(buf1, Wt[0][2], Bt[0][2], buf0, tilesM);
        k_gemm_wmma<64, 64, true, false><<<gTiles, 128, 0, stream>>>(buf0, Wt[0][3], Bt[0][3], z0, tilesM);
    }

    // ================= Path 1 (hidden = 128) =================
    {
        const int F = 128, SH = 7;
        int gTiles  = cdiv_i((long long)tilesM * (F / 16), 4);
        int gTiles4 = cdiv_i((long long)tilesM * (64 / 16), 4);
        int gInit   = cdiv_i((long long)N * F, TB);
        int gEdge   = cdiv_i((long long)E * (F / 4), TB);

        k_gemm_wmma<128, 128, false, false><<<gTiles, 128, 0, stream>>>(x, Wt[1][0], nullptr, buf0, tilesM);
        k_agg_init<<<gInit, TB, 0, stream>>>(buf0, Bt[1][0], dinv, buf1, N * F, SH);
        k_agg_edges<<<gEdge, TB, 0, stream>>>(buf0, src, dst, dinv, buf1, E * (F / 4), SH);
        k_gemm_wmma<128, 128, false, false><<<gTiles, 128, 0, stream>>>(buf1, Wt[1][1], nullptr, buf0, tilesM);
        k_agg_init<<<gInit, TB, 0, stream>>>(buf0, Bt[1][1], dinv, buf1, N * F, SH);
        k_agg_edges<<<gEdge, TB, 0, stream>>>(buf0, src, dst, dinv, buf1, E * (F / 4), SH);
        k_gemm_wmma<128, 128, true, true><<<gTiles, 128, 0, stream>>>(buf1, Wt[1][2], Bt[1][2], buf0, tilesM);
        k_gemm_wmma<128, 64, true, false><<<gTiles4, 128, 0, stream>>>(buf0, Wt[1][3], Bt[1][3], z1, tilesM);
    }

    // ---- combine paths + decode ----
    k_combine<<<cdiv_i((long long)N * 64, TB), TB, 0, stream>>>(z0, z1, zc, N * 64);
    k_decode<<<cdiv_i((long long)L * 16, TB), TB, 0, stream>>>(zc, sl, dl, out, L);
}